// LSTM_36137854828671
// MI455X (gfx1250) — compile-verified
//
#include <hip/hip_runtime.h>
#include <cstdint>
#include <cstddef>
#include <cmath>

#define BATCH   8
#define CH      256
#define HH      128
#define WW      128
#define TH      32          // output rows per tile
#define THREADS 256
#define TILES_PER_PLANE (HH / TH)

#define SY_ROWS (TH + 4)    // 36: y slab with halo 2
#define XS_ROWS (TH + 2)    // 34: x_stage with halo 1
#define XS_STRIDE (WW + 2)  // 130: zero-padded columns, no edge selects in conv2

// ---------------- CDNA5-specific helpers ----------------

// Async global->LDS copy, 16 bytes per active lane (ASYNCcnt-tracked).
// ISA 10.x / 15.18.3: GLOBAL_LOAD_ASYNC_TO_LDS_B128, vdst = LDS byte address.
__device__ __forceinline__ void async_ld_b128(void* lds_dst, const void* gsrc) {
    unsigned lds_addr = (unsigned)(uintptr_t)lds_dst;  // low 32 bits of generic LDS ptr == LDS offset
    asm volatile("global_load_async_to_lds_b128 %0, %1, off"
                 :: "v"(lds_addr), "v"(gsrc)
                 : "memory");
}

__device__ __forceinline__ void wait_async0() {
    asm volatile("s_wait_asynccnt 0" ::: "memory");
}

__device__ __forceinline__ float fast_exp2(float x) {
#if defined(__has_builtin) && __has_builtin(__builtin_amdgcn_exp2f)
    return __builtin_amdgcn_exp2f(x);        // v_exp_f32
#else
    return exp2f(x);
#endif
}

__device__ __forceinline__ float fast_rcp(float x) {
#if defined(__has_builtin) && __has_builtin(__builtin_amdgcn_rcpf)
    return __builtin_amdgcn_rcpf(x);         // v_rcp_f32
#else
    return 1.0f / x;
#endif
}

// ---------------- fused depthwise-gated kernel ----------------

__global__ __launch_bounds__(THREADS)
void dwgate_fused_kernel(const float* __restrict__ x,
                         const float* __restrict__ y,
                         const float* __restrict__ cw,
                         const float* __restrict__ cb,
                         float* __restrict__ out) {
    __shared__ __align__(16) float sy [SY_ROWS * WW];        // 18.0 KB, stride 128 (16B rows)
    __shared__ __align__(16) float sxs[XS_ROWS * XS_STRIDE]; // 17.3 KB, stride 130, zero-padded cols
    __shared__ __align__(16) float sfg[TH * WW];             // 16.0 KB

    const int tid   = threadIdx.x;
    const int blk   = blockIdx.x;
    const int plane = blk / TILES_PER_PLANE;            // b*C + c
    const int y0    = (blk % TILES_PER_PLANE) * TH;
    const int c     = plane & (CH - 1);

    // per-channel 3x3 filter + bias: block-uniform -> scalar loads
    const float* wp = cw + c * 9;
    const float w0 = wp[0], w1 = wp[1], w2 = wp[2];
    const float w3 = wp[3], w4 = wp[4], w5 = wp[5];
    const float w6 = wp[6], w7 = wp[7], w8 = wp[8];
    const float bias = cb[c];

    const size_t pbase = (size_t)plane * (size_t)(HH * WW);
    const float* yp = y + pbase;
    const float* xp = x + pbase;

    // ---- Phase 1: async-stage y rows [y0-2, y0+TH+2) into LDS; zero-fill OOB rows;
    //      zero the padded border columns of sxs ----
    for (int i = tid; i < SY_ROWS * (WW / 4); i += THREADS) {
        const int row = i >> 5;               // WW/4 == 32 float4 per row
        const int c4  = (i & 31) << 2;
        const int gr  = y0 - 2 + row;
        float* dst = &sy[row * WW + c4];
        if ((unsigned)gr < (unsigned)HH) {
            async_ld_b128(dst, yp + (size_t)gr * WW + c4);
        } else {
            *(float4*)dst = make_float4(0.f, 0.f, 0.f, 0.f);  // conv zero padding (rows)
        }
    }
    if (tid < XS_ROWS) {                      // zero-padded columns for conv2
        sxs[tid * XS_STRIDE]            = 0.f;
        sxs[tid * XS_STRIDE + (WW + 1)] = 0.f;
    }
    wait_async0();
    __syncthreads();

    // thread -> fixed column, rolling over rows
    const int  col  = tid & (WW - 1);
    const int  half = tid >> 7;               // 0 or 1
    const bool hasL = col > 0;
    const bool hasR = col < (WW - 1);

    // three weighted row-dots of sy row r at this column (top/mid/bottom roles)
    auto rowdots_sy = [&](int r, float& d0, float& d1, float& d2) {
        const float* s = &sy[r * WW + col];
        const float C = s[0];
        const float L = hasL ? s[-1] : 0.f;
        const float R = hasR ? s[ 1] : 0.f;
        d0 = fmaf(w0, L, fmaf(w1, C, w2 * R));
        d1 = fmaf(w3, L, fmaf(w4, C, w5 * R));
        d2 = fmaf(w6, L, fmaf(w7, C, w8 * R));
    };

    // same for padded x_stage rows (no edge selects)
    auto rowdots_xs = [&](int r, float& d0, float& d1, float& d2) {
        const float* s = &sxs[r * XS_STRIDE + col];
        const float L = s[0], C = s[1], R = s[2];
        d0 = fmaf(w0, L, fmaf(w1, C, w2 * R));
        d1 = fmaf(w3, L, fmaf(w4, C, w5 * R));
        d2 = fmaf(w6, L, fmaf(w7, C, w8 * R));
    };

    const float NL1 = -1.4426950408889634f;   // -log2(e)
    const float NL2 = -2.8853900817779268f;   // -2*log2(e)

    // ---- Phase 2: f_feat rows 0..33 (gRow = y0-1+j); gates; x_stage -> sxs ----
    {
        const int j0 = half * 17;             // 2 halves x 17 rows = 34
        float a0, b0, b1, u1, u2;
        rowdots_sy(j0,     a0, u1, u2);       // need only d0 of row j0
        rowdots_sy(j0 + 1, b0, b1, u2);       // need d0,d1 of row j0+1
        for (int k = 0; k < 17; ++k) {
            const int j  = j0 + k;
            const int gr = y0 - 1 + j;
            float n0, n1, n2;
            rowdots_sy(j + 2, n0, n1, n2);
            const float t = bias + (a0 + b1 + n2);       // f_feat

            // sigmoid & tanh from one v_exp_f32; rcp(inf)=0 gives correct saturation
            const float p   = fast_exp2(NL1 * t);                           // e^-t
            const float sig = fast_rcp(1.0f + p);
            const float th  = fmaf(2.0f, fast_rcp(fmaf(p, p, 1.0f)), -1.0f);

            float xs = 0.0f;                  // zero padding rows for conv2
            if ((unsigned)gr < (unsigned)HH) {
                xs = sig * (xp[(size_t)gr * WW + col] + th);
            }
            sxs[j * XS_STRIDE + col + 1] = xs;
            if (j >= 1 && j <= TH)
                sfg[(j - 1) * WW + col] = sig;

            a0 = b0; b0 = n0; b1 = n1;
        }
    }
    __syncthreads();

    // ---- Phase 3: conv2 on x_stage, tanh, gate, +2y, store ----
    {
        float* op = out + pbase + (size_t)y0 * WW;
        const int k0 = half * 16;             // 2 halves x 16 rows = 32
        float a0, b0, b1, u1, u2;
        rowdots_xs(k0,     a0, u1, u2);
        rowdots_xs(k0 + 1, b0, b1, u2);
        for (int k = 0; k < 16; ++k) {
            const int r = k0 + k;
            float n0, n1, n2;
            rowdots_xs(r + 2, n0, n1, n2);
            const float t = bias + (a0 + b1 + n2);

            const float p2  = fast_exp2(NL2 * t);                           // e^-2t
            const float th2 = fmaf(2.0f, fast_rcp(1.0f + p2), -1.0f);       // tanh(t)

            const float yv = sy[(r + 2) * WW + col];     // y at gRow, reused from LDS
            op[(size_t)r * WW + col] = fmaf(th2, sfg[r * WW + col], yv + yv);

            a0 = b0; b0 = n0; b1 = n1;
        }
    }
}

// ---------------- host-side launch ----------------

extern "C" void kernel_launch(void* const* d_in, const int* in_sizes, int n_in,
                              void* d_out, int out_size, void* d_ws, size_t ws_size,
                              hipStream_t stream) {
    (void)in_sizes; (void)n_in; (void)out_size; (void)d_ws; (void)ws_size;
    const float* x  = (const float*)d_in[0];
    const float* y  = (const float*)d_in[1];
    const float* cw = (const float*)d_in[2];
    const float* cb = (const float*)d_in[3];
    float* out = (float*)d_out;

    const int grid = BATCH * CH * TILES_PER_PLANE;   // 8192 workgroups
    dwgate_fused_kernel<<<grid, THREADS, 0, stream>>>(x, y, cw, cb, out);
}